// NCutsLoss_1297080124086
// MI455X (gfx1250) — compile-verified
//
#include <hip/hip_runtime.h>
#include <math.h>

#ifndef __has_builtin
#define __has_builtin(x) 0
#endif

// ---------------- problem constants (from reference) ----------------
#define NB    8
#define KB    20
#define XD    224
#define YD    224
#define CC    3
#define PADR  5
#define NPIX  (XD * YD)          // 50176
#define TILE  16
#define HALO  25                 // TILE + 5 + 4  (offsets dm,dn in [-5,4])
#define HSTR  26                 // LDS row stride
#define INV_SXSI (1.0f / (4.0f * 10.0f))  // 1/(SIGMA_X*SIGMA_I)

// workspace layout (floats): [0,32) per-n stats {S0,S1,S2,s2}; [32,192) accA; [192,352) accV
#define WS_STATS 0
#define WS_ACCA  32
#define WS_ACCV  192
#define WS_TOTAL 352

// ---------------- address-space helpers (value-preserving casts) ----------------
// Builtin wants  (AS1 int* gsrc, AS3 int* ldst, imm offset, imm cpol).
// Generic LDS pointer = {shared aperture, lds_offset[31:0]} -> low 32 bits are the AS(3) address.
typedef __attribute__((address_space(1))) int glob_i32;
typedef __attribute__((address_space(3))) int lds_i32;
static __device__ __forceinline__ glob_i32* as_glob(const void* p) {
  return (glob_i32*)(unsigned long long)p;
}
static __device__ __forceinline__ lds_i32* as_lds(const void* p) {
  return (lds_i32*)(unsigned)(unsigned long long)p;
}

// ---------------- CDNA5 async global->LDS copy (ASYNCcnt path) ----------------
#if __has_builtin(__builtin_amdgcn_global_load_async_to_lds_b32)
#define ASYNC_CP32(dst, src) \
  __builtin_amdgcn_global_load_async_to_lds_b32(as_glob(src), as_lds(dst), 0, 0)
#define HAVE_ASYNC_LDS 1
#else
#define ASYNC_CP32(dst, src) (*(dst) = *(src))
#define HAVE_ASYNC_LDS 0
#endif

#if __has_builtin(__builtin_amdgcn_s_wait_asynccnt)
#define ASYNC_WAIT0() __builtin_amdgcn_s_wait_asynccnt(0)
#elif HAVE_ASYNC_LDS
#define ASYNC_WAIT0() asm volatile("s_wait_asynccnt 0" ::: "memory")
#else
#define ASYNC_WAIT0() do {} while (0)
#endif

// ---------------- WMMA FP32 wave reduction ----------------
typedef float v2f __attribute__((ext_vector_type(2)));
typedef float v8f __attribute__((ext_vector_type(8)));

// Full-wave (32 lane) sum of v, returned valid in lanes 0..15 (we use lane 0).
// Uses V_WMMA_F32_16X16X4_F32 as a row-sum engine:
//   A (16x4): lanes 0-15 supply K=0,1 (v0,v1), lanes 16-31 supply K=2,3.
//   With v0=v1=v and B = all-ones -> D[m,n] = 2*(v_m + v_{m+16}) for every n.
//   Lane L<16:  sum(c[0..7]) = 2*sum_{m=0..7}(v_m+v_{m+16});  lane L+16: m=8..15 part.
static __device__ __forceinline__ float wave_sum32(float v) {
#if __has_builtin(__builtin_amdgcn_wmma_f32_16x16x4_f32)
  v2f a; a[0] = v;    a[1] = v;
  v2f b; b[0] = 1.0f; b[1] = 1.0f;
  v8f c = {0.f, 0.f, 0.f, 0.f, 0.f, 0.f, 0.f, 0.f};
  c = __builtin_amdgcn_wmma_f32_16x16x4_f32(false, a, false, b, (short)0, c,
                                            false, false);
  float t = c[0] + c[1] + c[2] + c[3] + c[4] + c[5] + c[6] + c[7];
  t += __shfl_down(t, 16, 32);
  return 0.5f * t;
#else
  for (int d = 16; d > 0; d >>= 1) v += __shfl_down(v, d, 32);
  return v;
#endif
}

// ---------------- kernel 0: zero workspace ----------------
__global__ void ncuts_zero(float* __restrict__ ws) {
  int t = blockIdx.x * 256 + threadIdx.x;
  if (t < WS_TOTAL) ws[t] = 0.0f;
}

// ---------------- kernel 1: per-image sums S[n][c], s2[n] ----------------
__global__ __launch_bounds__(256) void ncuts_stats(const float* __restrict__ x,
                                                   float* __restrict__ ws) {
  const int n = blockIdx.z;
  const int chunk = blockIdx.x;               // 28 chunks * 1792 pixels = 50176
  const float* xb = x + (size_t)n * CC * NPIX;
  const int per = NPIX / 28;
  const int base = chunk * per;
  float s0 = 0.f, s1 = 0.f, s2 = 0.f, qq = 0.f;
  for (int p = base + (int)threadIdx.x; p < base + per; p += 256) {
    float a = xb[p], b = xb[NPIX + p], c = xb[2 * NPIX + p];
    s0 += a; s1 += b; s2 += c;
    qq += a * a + b * b + c * c;
  }
  for (int d = 16; d > 0; d >>= 1) {
    s0 += __shfl_down(s0, d, 32);
    s1 += __shfl_down(s1, d, 32);
    s2 += __shfl_down(s2, d, 32);
    qq += __shfl_down(qq, d, 32);
  }
  if ((threadIdx.x & 31) == 0) {
    atomicAdd(&ws[WS_STATS + n * 4 + 0], s0);
    atomicAdd(&ws[WS_STATS + n * 4 + 1], s1);
    atomicAdd(&ws[WS_STATS + n * 4 + 2], s2);
    atomicAdd(&ws[WS_STATS + n * 4 + 3], qq);
  }
}

// ---------------- kernel 2: main tiled stencil + reductions ----------------
// One block per (n, 16x16 tile). x halo loaded once; seg halo double-buffered
// per k with async global->LDS loads overlapped with compute of the previous k.
__global__ __launch_bounds__(256) void ncuts_main(const float* __restrict__ seg,
                                                  const float* __restrict__ x,
                                                  float* __restrict__ ws) {
  __shared__ float xs[CC][HALO][HSTR];
  __shared__ float ssb[2][HALO][HSTR];

  const int n   = blockIdx.z;
  const int i0  = blockIdx.y * TILE;
  const int j0  = blockIdx.x * TILE;
  const int tid = threadIdx.x;
  const int tx  = tid & (TILE - 1);
  const int ty  = tid >> 4;

  const float* xb = x + (size_t)n * CC * NPIX;
  const float* sb = seg + (size_t)n * KB * NPIX;

  // ---- preamble: zero OOB halo slots, async-load x planes + seg(k=0) ----
  for (int e = tid; e < HALO * HALO; e += 256) {
    const int hy = e / HALO, hx = e - hy * HALO;
    const int gi = i0 - PADR + hy, gj = j0 - PADR + hx;
    if (gi >= 0 && gi < XD && gj >= 0 && gj < YD) {
      const int g = gi * YD + gj;
      ASYNC_CP32(&xs[0][hy][hx], xb + g);
      ASYNC_CP32(&xs[1][hy][hx], xb + NPIX + g);
      ASYNC_CP32(&xs[2][hy][hx], xb + 2 * NPIX + g);
      ASYNC_CP32(&ssb[0][hy][hx], sb + g);
    } else {
      xs[0][hy][hx] = 0.f; xs[1][hy][hx] = 0.f; xs[2][hy][hx] = 0.f;
      ssb[0][hy][hx] = 0.f; ssb[1][hy][hx] = 0.f;   // OOB slots stay 0 for every k
    }
  }
  ASYNC_WAIT0();
  __syncthreads();

  // ---- per-thread, k-independent precompute ----
  const int cy = ty + PADR, cx = tx + PADR;
  const float xc0 = xs[0][cy][cx], xc1 = xs[1][cy][cx], xc2 = xs[2][cy][cx];
  const float S0 = ws[WS_STATS + n * 4 + 0];
  const float S1 = ws[WS_STATS + n * 4 + 1];
  const float S2 = ws[WS_STATS + n * 4 + 2];
  const float s2 = ws[WS_STATS + n * 4 + 3];
  const float sq = xc0 * xc0 + xc1 * xc1 + xc2 * xc2;
  const float tempv =
      (float)NPIX * sq - 2.0f * (xc0 * S0 + xc1 * S1 + xc2 * S2) + s2;

  // 31 active offsets (d2 >= 25 within [-5,4]^2): weights in registers
  float w[31];
  {
    int idx = 0;
#pragma unroll
    for (int dm = -PADR; dm < PADR; ++dm) {
#pragma unroll
      for (int dn = -PADR; dn < PADR; ++dn) {
        const int d2 = dm * dm + dn * dn;
        if (d2 >= 25) {
          const float a0 = xc0 - xs[0][cy + dm][cx + dn];
          const float a1 = xc1 - xs[1][cy + dm][cx + dn];
          const float a2 = xc2 - xs[2][cy + dm][cx + dn];
          const float diff = __builtin_sqrtf(a0 * a0 + a1 * a1 + a2 * a2);
          w[idx++] = ((float)d2 * INV_SXSI) * diff;
        }
      }
    }
  }

  float* accA = ws + WS_ACCA;
  float* accV = ws + WS_ACCV;

  // ---- loop over k with double-buffered async seg halo ----
  for (int k = 0; k < KB; ++k) {
    if (k > 0) {                 // buffer k&1 was prefetched last iteration
      ASYNC_WAIT0();
      __syncthreads();           // all waves done reading buf[(k-1)&1] and have buf[k&1]
    }
    if (k < KB - 1) {            // prefetch k+1 into the other buffer (overlaps compute)
      const float* sk = sb + (size_t)(k + 1) * NPIX;
      float(*dst)[HSTR] = ssb[(k + 1) & 1];
      for (int e = tid; e < HALO * HALO; e += 256) {
        const int hy = e / HALO, hx = e - hy * HALO;
        const int gi = i0 - PADR + hy, gj = j0 - PADR + hx;
        if (gi >= 0 && gi < XD && gj >= 0 && gj < YD)
          ASYNC_CP32(&dst[hy][hx], sk + gi * YD + gj);
      }
    }

    // ---- compute: G = sum_o w[o]*seg_shift ; pa = seg*G ; pv = seg*temp ----
    float(*cur)[HSTR] = ssb[k & 1];
    const float sc = cur[cy][cx];
    float g = 0.f;
    {
      int idx = 0;
#pragma unroll
      for (int dm = -PADR; dm < PADR; ++dm) {
#pragma unroll
        for (int dn = -PADR; dn < PADR; ++dn) {
          const int d2 = dm * dm + dn * dn;
          if (d2 >= 25) g += w[idx++] * cur[cy + dm][cx + dn];
        }
      }
    }
    const float pa = sc * g;
    const float pv = sc * tempv;

    // wave-level WMMA reduction (EXEC all-ones here: no divergence in this phase)
    const float ra = wave_sum32(pa);
    const float rv = wave_sum32(pv);
    if ((tid & 31) == 0) {
      atomicAdd(&accA[n * KB + k], ra);
      atomicAdd(&accV[n * KB + k], rv);
    }
  }
}

// ---------------- kernel 3: loss = N*K - sum(accA/accV) ----------------
__global__ __launch_bounds__(256) void ncuts_final(const float* __restrict__ ws,
                                                   float* __restrict__ out) {
  __shared__ float red[256];
  const int t = threadIdx.x;
  float v = 0.f;
  if (t < NB * KB) v = ws[WS_ACCA + t] / ws[WS_ACCV + t];
  red[t] = v;
  __syncthreads();
  for (int s = 128; s > 0; s >>= 1) {
    if (t < s) red[t] += red[t + s];
    __syncthreads();
  }
  if (t == 0) out[0] = (float)(NB * KB) - red[0];
}

// ---------------- host-side launcher ----------------
extern "C" void kernel_launch(void* const* d_in, const int* in_sizes, int n_in,
                              void* d_out, int out_size, void* d_ws,
                              size_t ws_size, hipStream_t stream) {
  (void)in_sizes; (void)n_in; (void)out_size; (void)ws_size;
  const float* seg = (const float*)d_in[0];   // [8,20,224,224] f32
  const float* x   = (const float*)d_in[1];   // [8,3,224,224]  f32
  float* ws  = (float*)d_ws;
  float* out = (float*)d_out;

  ncuts_zero<<<2, 256, 0, stream>>>(ws);
  ncuts_stats<<<dim3(28, 1, NB), 256, 0, stream>>>(x, ws);
  ncuts_main<<<dim3(YD / TILE, XD / TILE, NB), 256, 0, stream>>>(seg, x, ws);
  ncuts_final<<<1, 256, 0, stream>>>(ws, out);
}